// EnvGen_61117384622468
// MI455X (gfx1250) — compile-verified
//
#include <hip/hip_runtime.h>
#include <hip/hip_bf16.h>
#include <math.h>

// ---------------------------------------------------------------------------
// Types for CDNA5 WMMA fragments
// ---------------------------------------------------------------------------
typedef float v2f __attribute__((ext_vector_type(2)));
typedef float v8f __attribute__((ext_vector_type(8)));

#define DIMF 128          // feature dim (D == O == 128)
#define ACT_NONE 0
#define ACT_RELU 1
#define ACT_SIGMOID 2

// ---------------------------------------------------------------------------
// Utility kernels
// ---------------------------------------------------------------------------
__global__ void fill_zero_kernel(float* __restrict__ p, unsigned n) {
    unsigned i = blockIdx.x * blockDim.x + threadIdx.x;
    if (i < n) p[i] = 0.0f;
}

// ---------------------------------------------------------------------------
// Degree accumulation (weighted + unweighted) over edge dst, then
// dinv = rsqrt(deg + 1)   (self-loop contributes weight 1 to both)
// ---------------------------------------------------------------------------
__global__ void deg_edges_kernel(const long long* __restrict__ dst,
                                 const float* __restrict__ ew,
                                 float* __restrict__ deg_es,
                                 float* __restrict__ deg_1,
                                 int E) {
    int e = blockIdx.x * blockDim.x + threadIdx.x;
    if (e >= E) return;
    int d = (int)dst[e];
    atomicAdd(&deg_es[d], ew[e]);
    atomicAdd(&deg_1[d], 1.0f);
}

__global__ void deg_finalize_kernel(float* __restrict__ deg_es,
                                    float* __restrict__ deg_1,
                                    int N) {
    int i = blockIdx.x * blockDim.x + threadIdx.x;
    if (i >= N) return;
    // self loop (+1). deg+1 > 0 always, matches where(deg>0, rsqrt, 0)
    deg_es[i] = rsqrtf(deg_es[i] + 1.0f);
    deg_1[i]  = rsqrtf(deg_1[i]  + 1.0f);
}

// ---------------------------------------------------------------------------
// Dense GEMM: out[M,128] = A[M,128] @ W[128,128]  (+bias, +activation)
// using V_WMMA_F32_16X16X4_F32 (exact fp32 matrix pipe).
// Grid: M/16 blocks of 256 threads (8 waves). Wave w computes column tile w.
// EXEC is all-ones throughout (WMMA requirement).
// ---------------------------------------------------------------------------
__global__ __launch_bounds__(256)
void gemm_wmma_f32_kernel(const float* __restrict__ A,
                          const float* __restrict__ W,
                          float* __restrict__ out,
                          const float* __restrict__ bias,  // may be null
                          int act) {
    const int wave = threadIdx.x >> 5;          // 0..7 -> column tile
    const int lane = threadIdx.x & 31;
    const int half = lane >> 4;                 // 0 or 1
    const int l16  = lane & 15;
    const int tile_m = blockIdx.x;

    // A fragment source: row = l16, contiguous float2 at k + 2*half
    const float* Arow = A + (size_t)(tile_m * 16 + l16) * DIMF;
    // B fragment source: column = l16 within this wave's 16-col tile
    const int ncol = wave * 16 + l16;

    v8f c = {0.f, 0.f, 0.f, 0.f, 0.f, 0.f, 0.f, 0.f};

#pragma unroll
    for (int k = 0; k < DIMF; k += 4) {
        // A 16x4 f32: lane<16 holds K=k..k+1, lane>=16 holds K=k+2..k+3
        v2f a = *(const v2f*)(Arow + k + 2 * half);
        // B 4x16 f32: same K split, one column per lane
        v2f b;
        b.x = W[(size_t)(k + 2 * half)     * DIMF + ncol];
        b.y = W[(size_t)(k + 2 * half + 1) * DIMF + ncol];
        c = __builtin_amdgcn_wmma_f32_16x16x4_f32(
                /*neg_a=*/false, a, /*neg_b=*/false, b,
                /*c_mod=*/(short)0, c, /*reuse_a=*/false, /*reuse_b=*/false);
    }

    // C/D layout: VGPR r -> row (r + 8*half), col = l16
    float* Od = out + (size_t)(tile_m * 16 + half * 8) * DIMF + ncol;
#pragma unroll
    for (int r = 0; r < 8; ++r) {
        float v = c[r];
        if (bias) v += bias[ncol];
        if (act == ACT_RELU)         v = fmaxf(v, 0.0f);
        else if (act == ACT_SIGMOID) v = 1.0f / (1.0f + expf(-v));
        Od[(size_t)r * DIMF] = v;
    }
}

// ---------------------------------------------------------------------------
// Sparse aggregation: out[dst] += h[src] * (dinv[src]*w*dinv[dst])
// One thread per (edge, feature). Gathers hit L2 (h is 25.6MB << 192MB).
// ---------------------------------------------------------------------------
__global__ void scatter_kernel(const long long* __restrict__ src,
                               const long long* __restrict__ dst,
                               const float* __restrict__ ew,   // null -> 1.0
                               const float* __restrict__ dinv,
                               const float* __restrict__ h,
                               float* __restrict__ out,
                               unsigned total) {               // E * 128
    unsigned idx = blockIdx.x * blockDim.x + threadIdx.x;
    if (idx >= total) return;
    unsigned e = idx >> 7;
    unsigned f = idx & (DIMF - 1);
    int s = (int)src[e];
    int d = (int)dst[e];
    float w = ew ? ew[e] : 1.0f;
    float norm = dinv[s] * w * dinv[d];
    atomicAdd(&out[(size_t)d * DIMF + f], h[(size_t)s * DIMF + f] * norm);
}

// ---------------------------------------------------------------------------
// GCN epilogue: agg = act(agg + h*dinv^2 (self loop) + bias) [+ time_emb[t]]
// ---------------------------------------------------------------------------
__global__ void gcn_epilogue_kernel(float* __restrict__ agg,
                                    const float* __restrict__ h,
                                    const float* __restrict__ dinv,
                                    const float* __restrict__ bias,
                                    const float* __restrict__ time_emb, // null if unused
                                    const int* __restrict__ t_ptr,
                                    int act, unsigned total) {          // N * 128
    unsigned idx = blockIdx.x * blockDim.x + threadIdx.x;
    if (idx >= total) return;
    unsigned i = idx >> 7;
    unsigned f = idx & (DIMF - 1);
    float di = dinv[i];
    float v = agg[idx] + h[idx] * di * di + bias[f];
    if (act == ACT_RELU)         v = fmaxf(v, 0.0f);
    else if (act == ACT_SIGMOID) v = 1.0f / (1.0f + expf(-v));
    if (time_emb) v += time_emb[(size_t)t_ptr[0] * DIMF + f];
    agg[idx] = v;
}

// ---------------------------------------------------------------------------
// Reparameterization: conf_z = eps * enc_std + enc_mean
// (deterministic hash-based Box-Muller; JAX PRNG stream not reproducible)
// ---------------------------------------------------------------------------
__device__ __forceinline__ float hash_u01(unsigned x) {
    x ^= x >> 17; x *= 0xed5ad4bbu;
    x ^= x >> 11; x *= 0xac4c1b51u;
    x ^= x >> 15; x *= 0x31848babu;
    x ^= x >> 14;
    return (float)(x >> 8) * (1.0f / 16777216.0f) + 1.0e-7f;
}

__global__ void confz_kernel(const float* __restrict__ enc_std,
                             const float* __restrict__ enc_mean,
                             float* __restrict__ out,
                             unsigned total) {
    unsigned idx = blockIdx.x * blockDim.x + threadIdx.x;
    if (idx >= total) return;
    float u1 = hash_u01(idx * 2654435761u + 0x9e3779b9u);
    float u2 = hash_u01(idx * 2246822519u + 0x85ebca6bu);
    float eps = sqrtf(-2.0f * logf(u1)) * cosf(6.28318530717958647f * u2);
    out[idx] = eps * enc_std[idx] + enc_mean[idx];
}

// ---------------------------------------------------------------------------
// Fused KL reduction: out[0] += scale * sum(kl)  (scale = 0.5/N)
// ---------------------------------------------------------------------------
__global__ void kl_reduce_kernel(const float* __restrict__ enc_std,
                                 const float* __restrict__ enc_mean,
                                 const float* __restrict__ prior_std,
                                 const float* __restrict__ prior_mean,
                                 float* __restrict__ out,
                                 float scale, unsigned total) {
    __shared__ float sdata[256];
    unsigned idx = blockIdx.x * blockDim.x + threadIdx.x;
    float kl = 0.0f;
    if (idx < total) {
        const float EPS = 1e-9f;
        float ps = prior_std[idx] + EPS;
        float es = enc_std[idx]   + EPS;
        float dm = enc_mean[idx] - prior_mean[idx];
        kl = 2.0f * logf(ps) - 2.0f * logf(es)
           + (es * es + dm * dm) / (ps * ps) - 1.0f;
    }
    sdata[threadIdx.x] = kl;
    __syncthreads();
#pragma unroll
    for (int s = 128; s > 0; s >>= 1) {
        if ((int)threadIdx.x < s) sdata[threadIdx.x] += sdata[threadIdx.x + s];
        __syncthreads();
    }
    if (threadIdx.x == 0) atomicAdd(out, sdata[0] * scale);
}

// ---------------------------------------------------------------------------
// Host-side orchestration
// ---------------------------------------------------------------------------
static inline unsigned cdiv(unsigned a, unsigned b) { return (a + b - 1) / b; }

extern "C" void kernel_launch(void* const* d_in, const int* in_sizes, int n_in,
                              void* d_out, int out_size, void* d_ws, size_t ws_size,
                              hipStream_t stream) {
    const long long* edge_index = (const long long*)d_in[0];
    const float*     x          = (const float*)d_in[1];
    const int*       t_ptr      = (const int*)d_in[2];
    const float*     edge_score = (const float*)d_in[3];
    const float*     W_enc   = (const float*)d_in[6];
    const float*     b_enc   = (const float*)d_in[7];
    const float*     W_mu    = (const float*)d_in[8];
    const float*     b_mu    = (const float*)d_in[9];
    const float*     W_std   = (const float*)d_in[10];
    const float*     b_std   = (const float*)d_in[11];
    const float*     W_prior = (const float*)d_in[12];
    const float*     b_prior = (const float*)d_in[13];
    const float*     W_pm    = (const float*)d_in[14];
    const float*     b_pm    = (const float*)d_in[15];
    const float*     W_ps    = (const float*)d_in[16];
    const float*     b_ps    = (const float*)d_in[17];
    const float*     time_emb = (const float*)d_in[18];

    const int N = in_sizes[1] / DIMF;      // 50000
    const int E = in_sizes[3];             // 800000
    const long long* src = edge_index;
    const long long* dst = edge_index + E;

    const unsigned NF = (unsigned)N * DIMF;
    const unsigned EF = (unsigned)E * DIMF;

    // workspace layout (floats)
    float* dinv_es   = (float*)d_ws;
    float* dinv_1    = dinv_es + N;
    float* h_tmp     = dinv_1 + N;          // N*128 : gemm staging / prior_mean
    float* bufA      = h_tmp + NF;          // N*128 : enc_t, then prior
    float* enc_mean  = bufA + NF;           // N*128
    float* enc_std   = enc_mean + NF;       // N*128
    float* prior_std = enc_std + NF;        // N*128

    float* kl_out  = (float*)d_out;         // scalar
    float* conf_z  = (float*)d_out + 1;     // N*128

    const dim3 blk(256);
    const dim3 grid_nf(cdiv(NF, 256));
    const dim3 grid_ef(cdiv(EF, 256));
    const dim3 grid_n(cdiv((unsigned)N, 256));
    const dim3 grid_e(cdiv((unsigned)E, 256));
    const dim3 grid_gemm((unsigned)N / 16); // N = 50000 = 3125*16 exactly

    // --- degrees & normalization -----------------------------------------
    fill_zero_kernel<<<grid_n, blk, 0, stream>>>(dinv_es, (unsigned)N);
    fill_zero_kernel<<<grid_n, blk, 0, stream>>>(dinv_1, (unsigned)N);
    deg_edges_kernel<<<grid_e, blk, 0, stream>>>(dst, edge_score, dinv_es, dinv_1, E);
    deg_finalize_kernel<<<grid_n, blk, 0, stream>>>(dinv_es, dinv_1, N);

    // --- GCN 1: enc_t = relu(gcn(x, W_enc, edge_score)) ------------------
    gemm_wmma_f32_kernel<<<grid_gemm, blk, 0, stream>>>(x, W_enc, h_tmp, nullptr, ACT_NONE);
    fill_zero_kernel<<<grid_nf, blk, 0, stream>>>(bufA, NF);
    scatter_kernel<<<grid_ef, blk, 0, stream>>>(src, dst, edge_score, dinv_es, h_tmp, bufA, EF);
    gcn_epilogue_kernel<<<grid_nf, blk, 0, stream>>>(bufA, h_tmp, dinv_es, b_enc,
                                                     nullptr, nullptr, ACT_RELU, NF);

    // --- GCN 2: enc_mean = gcn(enc_t, W_mu, ones) ------------------------
    gemm_wmma_f32_kernel<<<grid_gemm, blk, 0, stream>>>(bufA, W_mu, h_tmp, nullptr, ACT_NONE);
    fill_zero_kernel<<<grid_nf, blk, 0, stream>>>(enc_mean, NF);
    scatter_kernel<<<grid_ef, blk, 0, stream>>>(src, dst, nullptr, dinv_1, h_tmp, enc_mean, EF);
    gcn_epilogue_kernel<<<grid_nf, blk, 0, stream>>>(enc_mean, h_tmp, dinv_1, b_mu,
                                                     nullptr, nullptr, ACT_NONE, NF);

    // --- GCN 3: enc_std = sigmoid(gcn(enc_t, W_std, ones)) ---------------
    gemm_wmma_f32_kernel<<<grid_gemm, blk, 0, stream>>>(bufA, W_std, h_tmp, nullptr, ACT_NONE);
    fill_zero_kernel<<<grid_nf, blk, 0, stream>>>(enc_std, NF);
    scatter_kernel<<<grid_ef, blk, 0, stream>>>(src, dst, nullptr, dinv_1, h_tmp, enc_std, EF);
    gcn_epilogue_kernel<<<grid_nf, blk, 0, stream>>>(enc_std, h_tmp, dinv_1, b_std,
                                                     nullptr, nullptr, ACT_SIGMOID, NF);

    // --- conf_z = eps * enc_std + enc_mean -------------------------------
    confz_kernel<<<grid_nf, blk, 0, stream>>>(enc_std, enc_mean, conf_z, NF);

    // --- GCN 4: prior = relu(gcn(x, W_prior, ones)) + time_emb[t] --------
    gemm_wmma_f32_kernel<<<grid_gemm, blk, 0, stream>>>(x, W_prior, h_tmp, nullptr, ACT_NONE);
    fill_zero_kernel<<<grid_nf, blk, 0, stream>>>(bufA, NF);   // bufA now = prior
    scatter_kernel<<<grid_ef, blk, 0, stream>>>(src, dst, nullptr, dinv_1, h_tmp, bufA, EF);
    gcn_epilogue_kernel<<<grid_nf, blk, 0, stream>>>(bufA, h_tmp, dinv_1, b_prior,
                                                     time_emb, t_ptr, ACT_RELU, NF);

    // --- prior_mean / prior_std (dense, bias+act fused in GEMM) ----------
    gemm_wmma_f32_kernel<<<grid_gemm, blk, 0, stream>>>(bufA, W_pm, h_tmp, b_pm, ACT_NONE);
    gemm_wmma_f32_kernel<<<grid_gemm, blk, 0, stream>>>(bufA, W_ps, prior_std, b_ps, ACT_SIGMOID);

    // --- KL loss ----------------------------------------------------------
    fill_zero_kernel<<<dim3(1), dim3(1), 0, stream>>>(kl_out, 1u);
    kl_reduce_kernel<<<grid_nf, blk, 0, stream>>>(enc_std, enc_mean, prior_std, h_tmp,
                                                  kl_out, 0.5f / (float)N, NF);
}